// ImageModel_10024453669512
// MI455X (gfx1250) — compile-verified
//
#include <hip/hip_runtime.h>
#include <math.h>

typedef float v2f __attribute__((ext_vector_type(2)));
typedef float v8f __attribute__((ext_vector_type(8)));

// fast pow for y > 0 (matches jnp.power semantics incl. NaN for y < 0)
// v_log_f32 is log2; v_exp_f32 is exp2.
__device__ __forceinline__ float fpow(float y, float e) {
    return __builtin_amdgcn_exp2f(__builtin_amdgcn_logf(y) * e);
}

extern "C" __global__ void __launch_bounds__(256)
color_wmma_kernel(const float* __restrict__ x,
                  const float* __restrict__ S9,     // srgb 3x3 row-major
                  const float* __restrict__ gammap,
                  const float* __restrict__ brightp,
                  const float* __restrict__ offp,   // 3 floats
                  const float* __restrict__ huep,
                  float* __restrict__ out,
                  long long nPix)
{
    __shared__ float lds[8 * 64];                 // 64 floats (256B) per wave
    const int lane = threadIdx.x & 31;
    const int wib  = threadIdx.x >> 5;
    float* lbuf = &lds[wib * 64];

    const float invG   = 1.0f / gammap[0];
    const float bright = brightp[0];
    const float off0 = offp[0], off1 = offp[1], off2 = offp[2];
    const float th = huep[0];
    const float ch = cosf(th), sh = sinf(th);

    // combined = RGB_TO_YUV @ HUE(theta)
    const float Y00 = 0.299f,    Y01 = 0.587f,    Y02 = 0.114f;
    const float Y10 = -0.14713f, Y11 = -0.28886f, Y12 = 0.436f;
    const float Y20 = 0.615f,    Y21 = -0.51499f, Y22 = -0.10001f;
    float comb[9];
    comb[0] = Y00; comb[1] = Y01*ch + Y02*sh; comb[2] = -Y01*sh + Y02*ch;
    comb[3] = Y10; comb[4] = Y11*ch + Y12*sh; comb[5] = -Y11*sh + Y12*ch;
    comb[6] = Y20; comb[7] = Y21*ch + Y22*sh; comb[8] = -Y21*sh + Y22*ch;

    const int n  = lane & 15;     // column index within B / D
    const int hi = lane >> 4;     // 0 => K rows 0/1 ; 1 => K rows 2/3

    // B matrices in f32 16x16x4 layout: VGPR0 = K0|K2, VGPR1 = K1|K3
    v2f bS;
    bS.x = (n < 3) ? (hi ? S9[2*3 + n] : S9[0*3 + n]) : 0.0f;
    bS.y = (n < 3 && !hi) ? S9[1*3 + n] : 0.0f;      // K3 row = 0 (pad)
    v2f bC;
    bC.x = (n < 3) ? (hi ? comb[2*3 + n] : comb[0*3 + n]) : 0.0f;
    bC.y = (n < 3 && !hi) ? comb[1*3 + n] : 0.0f;

    const long long groups     = nPix >> 5;  // 32 pixels per wave-iteration
    const long long wavesTotal = (long long)gridDim.x * (blockDim.x >> 5);
    const long long wid        = (long long)blockIdx.x * (blockDim.x >> 5) + wib;

    for (long long g = wid; g < groups; g += wavesTotal) {
        const long long P = g << 5;
        #pragma unroll
        for (int h = 0; h < 2; ++h) {
            const long long base = P + h * 16;

            // ---- Build A for WMMA#1 directly from global (coalesced) ----
            // lanes 0-15: b64 load (r,g) of pixel `lane`; lanes 16-31: b32 load b
            v2f a;
            if (lane < 16) {
                const float* p = x + (base + lane) * 3;
                a.x = p[0]; a.y = p[1];
            } else {
                a.x = x[(base + (lane - 16)) * 3 + 2];
                a.y = 0.0f;
            }

            v8f zero8 = {};
            // D1[m][c] = sum_k x_m[k] * S[k][c]
            v8f d1 = __builtin_amdgcn_wmma_f32_16x16x4_f32(
                false, a, false, bS, (short)0, zero8, false, false);

            // ---- Transpose D1 (cols 0..2) back to A-layout via LDS ----
            if (n < 3) {
                #pragma unroll
                for (int i = 0; i < 8; ++i)
                    lbuf[(i + 8 * hi) * 4 + n] = d1[i];
            }
            __builtin_amdgcn_wave_barrier();
            asm volatile("s_wait_dscnt 0" ::: "memory");

            v2f a2;
            if (lane < 16) { a2.x = lbuf[lane * 4 + 0]; a2.y = lbuf[lane * 4 + 1]; }
            else           { a2.x = lbuf[(lane - 16) * 4 + 2]; a2.y = 0.0f; }

            // ---- Elementwise: pow(1/gamma), brightness, offset (packed) ----
            float vx = fpow(a2.x, invG) * bright + (hi ? off2 : off0);
            float vy = hi ? 0.0f : (fpow(a2.y, invG) * bright + off1);
            a2.x = vx; a2.y = vy;

            // D2[m][c] = sum_k y_m[k] * comb[k][c]
            v8f d2 = __builtin_amdgcn_wmma_f32_16x16x4_f32(
                false, a2, false, bC, (short)0, zero8, false, false);

            // ---- Repack D2 to contiguous [16][3] and store b128 ----
            __builtin_amdgcn_wave_barrier();
            if (n < 3) {
                #pragma unroll
                for (int i = 0; i < 8; ++i)
                    lbuf[(i + 8 * hi) * 3 + n] = d2[i];
            }
            __builtin_amdgcn_wave_barrier();
            asm volatile("s_wait_dscnt 0" ::: "memory");

            if (lane < 12) {
                const float4 v = *(const float4*)(lbuf + lane * 4);
                *(float4*)(out + base * 3 + lane * 4) = v;   // 192B / 16 px
            }
            __builtin_amdgcn_wave_barrier();
        }
    }

    // ---- Scalar tail for nPix % 32 (dead for this shape, kept for safety) ----
    const long long tailStart = groups << 5;
    const long long tid     = (long long)blockIdx.x * blockDim.x + threadIdx.x;
    const long long tstride = (long long)gridDim.x * blockDim.x;
    for (long long p = tailStart + tid; p < nPix; p += tstride) {
        const float* px = x + p * 3;
        float r = px[0], gq = px[1], b = px[2];
        float y0 = r * S9[0] + gq * S9[3] + b * S9[6];
        float y1 = r * S9[1] + gq * S9[4] + b * S9[7];
        float y2 = r * S9[2] + gq * S9[5] + b * S9[8];
        y0 = fpow(y0, invG) * bright + off0;
        y1 = fpow(y1, invG) * bright + off1;
        y2 = fpow(y2, invG) * bright + off2;
        out[p * 3 + 0] = y0 * comb[0] + y1 * comb[3] + y2 * comb[6];
        out[p * 3 + 1] = y0 * comb[1] + y1 * comb[4] + y2 * comb[7];
        out[p * 3 + 2] = y0 * comb[2] + y1 * comb[5] + y2 * comb[8];
    }
}

extern "C" void kernel_launch(void* const* d_in, const int* in_sizes, int n_in,
                              void* d_out, int out_size, void* d_ws, size_t ws_size,
                              hipStream_t stream) {
    const float* x      = (const float*)d_in[0];
    const float* S      = (const float*)d_in[1];
    const float* gamma  = (const float*)d_in[2];
    const float* bright = (const float*)d_in[3];
    const float* off    = (const float*)d_in[4];
    const float* hue    = (const float*)d_in[5];
    float* out = (float*)d_out;

    const long long nPix = (long long)in_sizes[0] / 3;
    const int threads = 256;                       // 8 waves per block
    const int wavesPerBlock = threads / 32;
    long long groups = nPix >> 5;
    long long need = (groups + wavesPerBlock - 1) / wavesPerBlock;
    if (need < 1) need = 1;
    int blocks = (int)(need < 4096 ? need : 4096); // grid-stride over 1M groups

    hipLaunchKernelGGL(color_wmma_kernel, dim3(blocks), dim3(threads), 0, stream,
                       x, S, gamma, bright, off, hue, out, nPix);
}